// GraphCorrector_89481348645711
// MI455X (gfx1250) — compile-verified
//
#include <hip/hip_runtime.h>

typedef float v2f __attribute__((ext_vector_type(2)));
typedef float v8f __attribute__((ext_vector_type(8)));

// ---------------- utility kernels ----------------

__global__ void zero_kernel(float* __restrict__ p, long n) {
    long i = (long)blockIdx.x * blockDim.x + threadIdx.x;
    if (i < n) p[i] = 0.0f;
}

__global__ void deg_kernel(const int* __restrict__ row, float* __restrict__ deg, int E) {
    int i = blockIdx.x * blockDim.x + threadIdx.x;
    if (i < E) atomicAdd(&deg[row[i]], 1.0f);
}

__global__ void invsqrt_kernel(float* __restrict__ deg, int n) {
    int i = blockIdx.x * blockDim.x + threadIdx.x;
    if (i < n) deg[i] = 1.0f / sqrtf(deg[i] + 1e-12f);
}

__global__ void norm_kernel(const int* __restrict__ row, const int* __restrict__ col,
                            const float* __restrict__ dis, float* __restrict__ nrm, int E) {
    int i = blockIdx.x * blockDim.x + threadIdx.x;
    if (i < E) nrm[i] = dis[row[i]] * dis[col[i]];
}

// ---------------- fp32 WMMA GEMM: Y[M,N] = (relu?)X[M,K] @ W[K,N] ----------------
// One wave computes a 16(M) x N tile using V_WMMA_F32_16X16X4_F32.
// A 16x4 layout: lane (m = lane&15) holds K = 2*(lane>>4) + {0,1}
// B 4x16 layout: lane (n = lane&15) holds K = 2*(lane>>4) + {0,1}
// C/D 16x16:     VGPR r holds row r + 8*(lane>>4), col = lane&15

template<int K, int N, bool RELU_IN>
__global__ void gemm_wmma(const float* __restrict__ X, const float* __restrict__ W,
                          float* __restrict__ Y, int M) {
    constexpr int NT = (N + 15) / 16;
    const int lane = threadIdx.x & 31;
    const int wave = threadIdx.x >> 5;
    const int tile = blockIdx.x * (blockDim.x >> 5) + wave;
    const int m0 = tile * 16;
    if (m0 >= M) return;                 // whole wave exits -> EXEC stays all-1s

    const int half = lane >> 4;          // 0 or 1
    const int mr   = lane & 15;
    const float* __restrict__ xrow = X + (long)(m0 + mr) * K + half * 2;

    v8f acc[NT] = {};

    for (int kk = 0; kk < K; kk += 4) {
        v2f a;
        a.x = xrow[kk];
        a.y = xrow[kk + 1];
        if (RELU_IN) { a.x = fmaxf(a.x, 0.0f); a.y = fmaxf(a.y, 0.0f); }
        const int kA = kk + half * 2;
#pragma unroll
        for (int t = 0; t < NT; ++t) {
            const int n = t * 16 + mr;
            v2f b; b.x = 0.0f; b.y = 0.0f;
            if (n < N) {
                b.x = W[(long)kA * N + n];
                b.y = W[(long)(kA + 1) * N + n];
            }
            acc[t] = __builtin_amdgcn_wmma_f32_16x16x4_f32(
                false, a, false, b, (short)0, acc[t], false, false);
        }
    }

#pragma unroll
    for (int t = 0; t < NT; ++t) {
        const int n = t * 16 + mr;
        if (n < N) {
#pragma unroll
            for (int r = 0; r < 8; ++r) {
                const int m = m0 + r + half * 8;
                Y[(long)m * N + n] = acc[t][r];
            }
        }
    }
}

// ---------------- edge scatter: dst[row] += src[col] * norm ----------------
// One thread handles 4 features (float4 gather) of one edge -> amortizes
// the row/col/norm index loads 4x; all traffic is L2-resident.

template<int F>
__global__ void scatter_kernel(const float* __restrict__ src, const int* __restrict__ row,
                               const int* __restrict__ col, const float* __restrict__ nrm,
                               float* __restrict__ dst, int E) {
    constexpr int C4 = F / 4;
    long i = (long)blockIdx.x * blockDim.x + threadIdx.x;
    if (i >= (long)E * C4) return;
    const int e = (int)(i / C4);
    const int c = (int)(i % C4);
    const int r  = row[e];
    const int cl = col[e];
    const float nm = nrm[e];
    const float4 v = *(const float4*)(src + (long)cl * F + c * 4);
    float* d = dst + (long)r * F + c * 4;
    atomicAdd(d + 0, v.x * nm);
    atomicAdd(d + 1, v.y * nm);
    atomicAdd(d + 2, v.z * nm);
    atomicAdd(d + 3, v.w * nm);
}

// ---------------- launch ----------------

extern "C" void kernel_launch(void* const* d_in, const int* in_sizes, int n_in,
                              void* d_out, int out_size, void* d_ws, size_t ws_size,
                              hipStream_t stream) {
    const float* x    = (const float*)d_in[0];
    const int*   ei   = (const int*)  d_in[1];
    const float* W1   = (const float*)d_in[2];
    const float* W2   = (const float*)d_in[3];
    const float* Wout = (const float*)d_in[4];
    float* out = (float*)d_out;

    const int NN = in_sizes[0] / 32;   // 100000
    const int E  = in_sizes[1] / 2;    // 1600000
    const int* row = ei;
    const int* col = ei + E;

    // workspace carve-out (256B-aligned chunks)
    float* ws = (float*)d_ws;
    size_t off = 0;
    auto carve = [&](size_t nflt) {
        float* p = ws + off;
        off += (nflt + 63) & ~(size_t)63;
        return p;
    };
    float* deg  = carve((size_t)NN);
    float* nrm  = carve((size_t)E);
    float* bufA = carve((size_t)NN * 48);
    float* bufB = carve((size_t)NN * 48);
    float* bufC = carve((size_t)NN * 8);

    auto cdiv = [](long a, long b) { return (int)((a + b - 1) / b); };
    const int T = 256;

    // norm = deg^-1/2[row] * deg^-1/2[col]
    zero_kernel   <<<cdiv(NN, T), T, 0, stream>>>(deg, NN);
    deg_kernel    <<<cdiv(E,  T), T, 0, stream>>>(row, deg, E);
    invsqrt_kernel<<<cdiv(NN, T), T, 0, stream>>>(deg, NN);
    norm_kernel   <<<cdiv(E,  T), T, 0, stream>>>(row, col, deg, nrm, E);

    const int gemm_blocks = cdiv((NN + 15) / 16, T / 32);

    // layer 1: bufB = scatter(x @ W1)
    gemm_wmma<32, 48, false><<<gemm_blocks, T, 0, stream>>>(x, W1, bufA, NN);
    zero_kernel<<<cdiv((long)NN * 48, T), T, 0, stream>>>(bufB, (long)NN * 48);
    scatter_kernel<48><<<cdiv((long)E * 12, T), T, 0, stream>>>(bufA, row, col, nrm, bufB, E);

    // layer 2: bufB = scatter(relu(bufB) @ W2)   (gemm reads bufB before it is re-zeroed)
    gemm_wmma<48, 48, true><<<gemm_blocks, T, 0, stream>>>(bufB, W2, bufA, NN);
    zero_kernel<<<cdiv((long)NN * 48, T), T, 0, stream>>>(bufB, (long)NN * 48);
    scatter_kernel<48><<<cdiv((long)E * 12, T), T, 0, stream>>>(bufA, row, col, nrm, bufB, E);

    // output layer: out = scatter(relu(bufB) @ W_out)
    gemm_wmma<48, 8, true><<<gemm_blocks, T, 0, stream>>>(bufB, Wout, bufC, NN);
    zero_kernel<<<cdiv((long)NN * 8, T), T, 0, stream>>>(out, (long)NN * 8);
    scatter_kernel<8><<<cdiv((long)E * 2, T), T, 0, stream>>>(bufC, row, col, nrm, out, E);
}